// QueryAndGroupDeform_85323820302743
// MI455X (gfx1250) — compile-verified
//
#include <hip/hip_runtime.h>
#include <hip/hip_bf16.h>

#define NSAMPLE 32
#define CFEAT   64

typedef float v2f __attribute__((ext_vector_type(2)));
typedef float v8f __attribute__((ext_vector_type(8)));

__global__ void bqd_prep_points(const float* __restrict__ xyz,
                                float4* __restrict__ pts4, int ntot) {
    int i = blockIdx.x * blockDim.x + threadIdx.x;
    if (i >= ntot) return;
    float x = xyz[3 * i + 0], y = xyz[3 * i + 1], z = xyz[3 * i + 2];
    pts4[i] = make_float4(x, y, z, x * x + y * y + z * z);
}

__global__ void bqd_prep_queries(const float* __restrict__ nxyz,
                                 const float* __restrict__ nr,
                                 const float* __restrict__ tdec,
                                 float4* __restrict__ qa4, int mtot) {
    int i = blockIdx.x * blockDim.x + threadIdx.x;
    if (i >= mtot) return;
    float x = nxyz[3 * i + 0], y = nxyz[3 * i + 1], z = nxyz[3 * i + 2];
    float real_t = 0.1f * tdec[0];
    float re = nr[i] + real_t * 5.0f;
    qa4[i] = make_float4(x, y, z, re * re - (x * x + y * y + z * z));
}

// Balanced select tree over NAMED scalars (no alloca -> no LDS promotion).
__device__ __forceinline__ uint bqd_sel8(uint ls, uint b0, uint b1, uint b2, uint b3,
                                         uint b4, uint b5, uint b6, uint b7) {
    uint lo = (ls < 2u) ? ((ls == 0u) ? b0 : b1) : ((ls == 2u) ? b2 : b3);
    uint hv = (ls < 6u) ? ((ls == 4u) ? b4 : b5) : ((ls == 6u) ? b6 : b7);
    return (ls < 4u) ? lo : hv;
}

// Owner-lane m (0..15) extracts its 16 column bits; lanes >=16 get 0 so the
// append loop self-disables.
__device__ __forceinline__ uint bqd_row_bits(int lane, uint b0, uint b1, uint b2, uint b3,
                                             uint b4, uint b5, uint b6, uint b7) {
    uint bm = bqd_sel8((uint)(lane & 7), b0, b1, b2, b3, b4, b5, b6, b7);
    uint bits = (lane < 8) ? (bm & 0xFFFFu) : (bm >> 16);
    return (lane < 16) ? bits : 0u;
}

__device__ __forceinline__ void bqd_append(uint bits, int base, int* slot,
                                           int& cnt, int& first) {
    while (bits && cnt < NSAMPLE) {
        int bpos = __ffs((int)bits) - 1;
        int gi = base + bpos;
        if (cnt == 0) first = gi;
        slot[cnt] = gi;
        ++cnt;
        bits &= bits - 1;
    }
}

__global__ void __launch_bounds__(128)
bqd_main(const int* __restrict__ xyz_cnt,
         const int* __restrict__ new_cnt,
         const float* __restrict__ new_r,
         const float* __restrict__ features,
         const float* __restrict__ tdec,
         const float4* __restrict__ pts4,
         const float4* __restrict__ qa4,
         float* __restrict__ out_nf,
         float* __restrict__ out_w,
         int* __restrict__ out_idx,
         int btot, int ntot, int mtot) {
    const int lane = threadIdx.x & 31;
    const int wave = threadIdx.x >> 5;
    const bool hi  = lane >= 16;

    __shared__ int s_idx[4][16][NSAMPLE];
    __shared__ int s_cnt[4][16];

    int q0 = (blockIdx.x * (blockDim.x >> 5) + wave) * 16;
    if (q0 > mtot - 16) q0 = mtot - 16;

    // batch of this query tile
    int b = 0;
    {
        int acc = 0;
        for (int i = 0; i < btot; ++i) {
            int c = new_cnt[i];
            if (q0 >= acc + c) { acc += c; b++; } else break;
        }
    }
    int poff = 0, pcnt = 0;
    {
        int acc = 0;
        for (int i = 0; i < btot; ++i) {
            int c = xyz_cnt[i];
            if (i == b) { poff = acc; pcnt = c; }
            acc += c;
        }
    }

    const int qm = q0 + (lane & 15);
    const float4 Q = qa4[qm];
    const float r_orig = new_r[qm];

    // A fragment (16x4 f32): rows = (-2qx,-2qy,-2qz,1)
    v2f afrag;
    afrag.x = hi ? (-2.0f * Q.z) : (-2.0f * Q.x);
    afrag.y = hi ? 1.0f          : (-2.0f * Q.y);

    // per-D-VGPR row threshold thr = r_e^2 - |q|^2 (rows v / v+8)
    float thr[8];
#pragma unroll
    for (int v = 0; v < 8; ++v) thr[v] = __shfl(Q.w, hi ? (v + 8) : v, 32);

    const float4* P4 = pts4 + poff;
    const v8f czf = {0.f, 0.f, 0.f, 0.f, 0.f, 0.f, 0.f, 0.f};
    int* slot = &s_idx[wave][lane & 15][0];

    int cnt = 0, first = 0;
    const float4* pptr = P4 + (lane & 15);   // per-lane base; chunks advance by 32 pts
    const int nchunk = pcnt >> 5;            // full 32-point chunks

    for (int c = 0; c < nchunk; ++c) {
        const int n0 = c << 5;

        __builtin_prefetch((const void*)(pptr + n0 + 128), 0, 1);

        const float4 PA = pptr[n0];
        const float4 PB = pptr[n0 + 16];

        // B fragments (4x16): rows = (px,py,pz,|p|^2)
        v2f bfA, bfB;
        bfA.x = hi ? PA.z : PA.x;
        bfA.y = hi ? PA.w : PA.y;
        bfB.x = hi ? PB.z : PB.x;
        bfB.y = hi ? PB.w : PB.y;

        // D[m][n] = -2 q_m . p_n + |p_n|^2
        v8f dA = __builtin_amdgcn_wmma_f32_16x16x4_f32(
            false, afrag, false, bfA, (short)0, czf, false, false);
        v8f dB = __builtin_amdgcn_wmma_f32_16x16x4_f32(
            false, afrag, false, bfB, (short)0, czf, false, false);

        const uint a0 = __builtin_amdgcn_ballot_w32(dA[0] < thr[0]);
        const uint a1 = __builtin_amdgcn_ballot_w32(dA[1] < thr[1]);
        const uint a2 = __builtin_amdgcn_ballot_w32(dA[2] < thr[2]);
        const uint a3 = __builtin_amdgcn_ballot_w32(dA[3] < thr[3]);
        const uint a4 = __builtin_amdgcn_ballot_w32(dA[4] < thr[4]);
        const uint a5 = __builtin_amdgcn_ballot_w32(dA[5] < thr[5]);
        const uint a6 = __builtin_amdgcn_ballot_w32(dA[6] < thr[6]);
        const uint a7 = __builtin_amdgcn_ballot_w32(dA[7] < thr[7]);
        const uint b0 = __builtin_amdgcn_ballot_w32(dB[0] < thr[0]);
        const uint b1 = __builtin_amdgcn_ballot_w32(dB[1] < thr[1]);
        const uint b2 = __builtin_amdgcn_ballot_w32(dB[2] < thr[2]);
        const uint b3 = __builtin_amdgcn_ballot_w32(dB[3] < thr[3]);
        const uint b4 = __builtin_amdgcn_ballot_w32(dB[4] < thr[4]);
        const uint b5 = __builtin_amdgcn_ballot_w32(dB[5] < thr[5]);
        const uint b6 = __builtin_amdgcn_ballot_w32(dB[6] < thr[6]);
        const uint b7 = __builtin_amdgcn_ballot_w32(dB[7] < thr[7]);

        bqd_append(bqd_row_bits(lane, a0, a1, a2, a3, a4, a5, a6, a7), n0,      slot, cnt, first);
        bqd_append(bqd_row_bits(lane, b0, b1, b2, b3, b4, b5, b6, b7), n0 + 16, slot, cnt, first);

        uint done = __builtin_amdgcn_ballot_w32((lane < 16) ? (cnt >= NSAMPLE) : true);
        if (done == 0xFFFFFFFFu) break;
    }

    // Tail chunk (pcnt % 32 leftover). Safe after early exit: appends no-op
    // once every query row is full.
    const int ntail = pcnt - (nchunk << 5);
    if (ntail > 0) {
        const int n0 = nchunk << 5;
        const int pa = n0 + (lane & 15);
        const int pb = pa + 16;
        const int lclamp = pcnt - 1;
        const float4 PA = P4[min(pa, lclamp)];
        const float4 PB = P4[min(pb, lclamp)];
        v2f bfA, bfB;
        bfA.x = hi ? PA.z : PA.x;
        bfA.y = hi ? PA.w : PA.y;
        bfB.x = hi ? PB.z : PB.x;
        bfB.y = hi ? PB.w : PB.y;
        if (hi && pa >= pcnt) bfA.y = 3.0e38f;   // force OOB columns out of every ball
        if (hi && pb >= pcnt) bfB.y = 3.0e38f;

        v8f dA = __builtin_amdgcn_wmma_f32_16x16x4_f32(
            false, afrag, false, bfA, (short)0, czf, false, false);
        v8f dB = __builtin_amdgcn_wmma_f32_16x16x4_f32(
            false, afrag, false, bfB, (short)0, czf, false, false);

        const uint a0 = __builtin_amdgcn_ballot_w32(dA[0] < thr[0]);
        const uint a1 = __builtin_amdgcn_ballot_w32(dA[1] < thr[1]);
        const uint a2 = __builtin_amdgcn_ballot_w32(dA[2] < thr[2]);
        const uint a3 = __builtin_amdgcn_ballot_w32(dA[3] < thr[3]);
        const uint a4 = __builtin_amdgcn_ballot_w32(dA[4] < thr[4]);
        const uint a5 = __builtin_amdgcn_ballot_w32(dA[5] < thr[5]);
        const uint a6 = __builtin_amdgcn_ballot_w32(dA[6] < thr[6]);
        const uint a7 = __builtin_amdgcn_ballot_w32(dA[7] < thr[7]);
        const uint b0 = __builtin_amdgcn_ballot_w32(dB[0] < thr[0]);
        const uint b1 = __builtin_amdgcn_ballot_w32(dB[1] < thr[1]);
        const uint b2 = __builtin_amdgcn_ballot_w32(dB[2] < thr[2]);
        const uint b3 = __builtin_amdgcn_ballot_w32(dB[3] < thr[3]);
        const uint b4 = __builtin_amdgcn_ballot_w32(dB[4] < thr[4]);
        const uint b5 = __builtin_amdgcn_ballot_w32(dB[5] < thr[5]);
        const uint b6 = __builtin_amdgcn_ballot_w32(dB[6] < thr[6]);
        const uint b7 = __builtin_amdgcn_ballot_w32(dB[7] < thr[7]);

        bqd_append(bqd_row_bits(lane, a0, a1, a2, a3, a4, a5, a6, a7), n0,      slot, cnt, first);
        bqd_append(bqd_row_bits(lane, b0, b1, b2, b3, b4, b5, b6, b7), n0 + 16, slot, cnt, first);
    }

    if (lane < 16) {
        int pad = (cnt == 0) ? 0 : first;
        for (int s = cnt; s < NSAMPLE; ++s) slot[s] = pad;
        s_cnt[wave][lane] = cnt;
    }
    __syncthreads();

    // Phase B: 32 lanes = 32 samples -> every channel store is 128B contiguous
    const float real_t = 0.1f * tdec[0];
    for (int mq = 0; mq < 16; ++mq) {
        const int qg = q0 + mq;
        const bool emptyq = (s_cnt[wave][mq] == 0);
        const int li = s_idx[wave][mq][lane];

        const float qx = __shfl(Q.x, mq, 32);
        const float qy = __shfl(Q.y, mq, 32);
        const float qz = __shfl(Q.z, mq, 32);
        const float rq = __shfl(r_orig, mq, 32);

        const float4 P = P4[li];
        const float cx = emptyq ? 0.0f : P.x - qx;
        const float cy = emptyq ? 0.0f : P.y - qy;
        const float cz = emptyq ? 0.0f : P.z - qz;
        const float dist = sqrtf(cx * cx + cy * cy + cz * cz);
        const float w = 1.0f / (1.0f + __expf((dist - rq) / real_t));

        float* nf = out_nf + (size_t)qg * (3 + CFEAT) * NSAMPLE;
        nf[0 * NSAMPLE + lane] = cx;
        nf[1 * NSAMPLE + lane] = cy;
        nf[2 * NSAMPLE + lane] = cz;

        const float4* frow = (const float4*)(features + (size_t)(poff + li) * CFEAT);
#pragma unroll
        for (int c4 = 0; c4 < CFEAT / 4; ++c4) {
            float4 f = frow[c4];
            if (emptyq) { f.x = 0.f; f.y = 0.f; f.z = 0.f; f.w = 0.f; }
            nf[(3 + c4 * 4 + 0) * NSAMPLE + lane] = f.x;
            nf[(3 + c4 * 4 + 1) * NSAMPLE + lane] = f.y;
            nf[(3 + c4 * 4 + 2) * NSAMPLE + lane] = f.z;
            nf[(3 + c4 * 4 + 3) * NSAMPLE + lane] = f.w;
        }
        out_w[(size_t)qg * NSAMPLE + lane] = w;
        out_idx[(size_t)qg * NSAMPLE + lane] = li;
    }
}

extern "C" void kernel_launch(void* const* d_in, const int* in_sizes, int n_in,
                              void* d_out, int out_size, void* d_ws, size_t ws_size,
                              hipStream_t stream) {
    const float* xyz      = (const float*)d_in[0];
    const int*   xyz_cnt  = (const int*)d_in[1];
    const float* new_xyz  = (const float*)d_in[2];
    const float* new_r    = (const float*)d_in[3];
    const int*   new_cnt  = (const int*)d_in[4];
    const float* features = (const float*)d_in[5];
    const float* tdec     = (const float*)d_in[6];

    const int ntot = in_sizes[0] / 3;
    const int mtot = in_sizes[2] / 3;
    const int btot = in_sizes[1];

    float4* pts4 = (float4*)d_ws;
    float4* qa4  = (float4*)((char*)d_ws + (size_t)ntot * sizeof(float4));

    float* out_nf = (float*)d_out;
    float* out_w  = out_nf + (size_t)mtot * (3 + CFEAT) * NSAMPLE;
    int*   out_ix = (int*)(out_w + (size_t)mtot * NSAMPLE);

    bqd_prep_points<<<(ntot + 255) / 256, 256, 0, stream>>>(xyz, pts4, ntot);
    bqd_prep_queries<<<(mtot + 255) / 256, 256, 0, stream>>>(new_xyz, new_r, tdec, qa4, mtot);

    const int qper = 4 * 16;  // 4 waves x 16 queries per block
    bqd_main<<<(mtot + qper - 1) / qper, 128, 0, stream>>>(
        xyz_cnt, new_cnt, new_r, features, tdec, pts4, qa4,
        out_nf, out_w, out_ix, btot, ntot, mtot);
}